// MGCL_42932493091122
// MI455X (gfx1250) — compile-verified
//
#include <hip/hip_runtime.h>

#define NUM_USER 50000
#define NUM_ITEM 50000
#define NN (NUM_USER + NUM_ITEM)
#define FDIM 192          // 3 stacks x 64 fused columns
#define NEDGE 2000000
#define KC 64             // K-chunk staged in LDS

typedef __attribute__((ext_vector_type(2))) float v2f;
typedef __attribute__((ext_vector_type(8))) float v8f;

// ---------- degree / normalization ----------
__global__ void k_fill1(float* __restrict__ deg, int n) {
    int i = blockIdx.x * blockDim.x + threadIdx.x;
    if (i < n) deg[i] = 1.0f;                       // self-loop contribution
}

__global__ void k_deg_edges(const int* __restrict__ edges, float* __restrict__ deg, int e) {
    int i = blockIdx.x * blockDim.x + threadIdx.x;
    if (i >= e) return;
    int u  = edges[2 * i];
    int it = edges[2 * i + 1] + NUM_USER;
    atomicAdd(deg + u, 1.0f);
    atomicAdd(deg + it, 1.0f);
}

__global__ void k_rsqrt(float* __restrict__ d, int n) {
    int i = blockIdx.x * blockDim.x + threadIdx.x;
    if (i < n) d[i] = rsqrtf(d[i]);                 // deg >= 1 always
}

// ---------- X0 assembly ----------
__global__ void k_init_users(const float* __restrict__ ug, const float* __restrict__ uv,
                             const float* __restrict__ ut, float* __restrict__ X) {
    int idx = blockIdx.x * blockDim.x + threadIdx.x;    // NUM_USER*64 threads
    if (idx >= NUM_USER * 64) return;
    int n = idx >> 6, j = idx & 63;
    size_t base = (size_t)n * FDIM;
    X[base + j]        = ug[idx];
    X[base + 64 + j]   = uv[idx];
    X[base + 128 + j]  = ut[idx];
}

__global__ void k_init_items_g(const float* __restrict__ ie, float* __restrict__ X) {
    int idx = blockIdx.x * blockDim.x + threadIdx.x;    // NUM_ITEM*64 threads
    if (idx >= NUM_ITEM * 64) return;
    int n = idx >> 6, j = idx & 63;
    X[(size_t)(NUM_USER + n) * FDIM + j] = ie[idx];
}

// ---------- WMMA GEMM:  C[Mx64] = A[MxK] @ W[64xK]^T (+bias) ----------
// one wave computes a 16x64 tile; block = 8 waves = 128 rows.
// W chunks are double-buffered in LDS via gfx1250 async-to-LDS loads.
__global__ void k_gemm16x64(const float* __restrict__ A, int lda,
                            const float* __restrict__ W, int K,
                            const float* __restrict__ bias,
                            float* __restrict__ C, int ldc, int M) {
    __shared__ float wbuf[2][64 * KC];                  // 2 x 16KB
    const int tid  = threadIdx.x;
    const int lane = tid & 31;
    const int wave = tid >> 5;
    const int m0 = (blockIdx.x * 8 + wave) * 16;
    const bool active = (m0 < M);                       // wave-uniform
    const int r = lane & 15;
    const int half = lane >> 4;                         // lanes 16-31 hold K=2,3 of each group

    v8f acc0 = {}, acc1 = {}, acc2 = {}, acc3 = {};

    // ---- async staging of one 64(N) x KC(K) chunk of W into LDS, B-friendly [n][kk] layout
    auto stage = [&](int bufsel, int kc0) {
#pragma unroll
        for (int j = 0; j < 4; ++j) {                   // 1024 b128 xfers / 256 threads
            int flat4 = tid + j * 256;
            int n  = flat4 >> 4;                        // 0..63
            int kk = (flat4 & 15) << 2;                 // 0..60 step 4
            unsigned long long g =
                (unsigned long long)(size_t)(W + (size_t)n * K + kc0 + kk);
            unsigned l = (unsigned)(size_t)(&wbuf[bufsel][n * KC + kk]);
            asm volatile("global_load_async_to_lds_b128 %0, %1, off"
                         :: "v"(l), "v"(g) : "memory");
        }
    };

    auto compute = [&](int bufsel, int kc0) {
        if (!active) return;                            // wave-uniform; EXEC stays all-1
        // bulk-issue A loads for the whole chunk (pipelined against WMMA)
        v2f a[16];
        const float* Ar = A + (size_t)(m0 + r) * lda + kc0 + 2 * half;
#pragma unroll
        for (int g = 0; g < 16; ++g) a[g] = *(const v2f*)(Ar + 4 * g);
        const float* b0p = &wbuf[bufsel][(r +  0) * KC + 2 * half];
        const float* b1p = &wbuf[bufsel][(r + 16) * KC + 2 * half];
        const float* b2p = &wbuf[bufsel][(r + 32) * KC + 2 * half];
        const float* b3p = &wbuf[bufsel][(r + 48) * KC + 2 * half];
#pragma unroll
        for (int g = 0; g < 16; ++g) {
            v2f b0 = *(const v2f*)(b0p + 4 * g);
            v2f b1 = *(const v2f*)(b1p + 4 * g);
            v2f b2 = *(const v2f*)(b2p + 4 * g);
            v2f b3 = *(const v2f*)(b3p + 4 * g);
            acc0 = __builtin_amdgcn_wmma_f32_16x16x4_f32(false, a[g], false, b0, (short)0, acc0, false, false);
            acc1 = __builtin_amdgcn_wmma_f32_16x16x4_f32(false, a[g], false, b1, (short)0, acc1, false, false);
            acc2 = __builtin_amdgcn_wmma_f32_16x16x4_f32(false, a[g], false, b2, (short)0, acc2, false, false);
            acc3 = __builtin_amdgcn_wmma_f32_16x16x4_f32(false, a[g], false, b3, (short)0, acc3, false, false);
        }
    };

    const int nch = K / KC;                             // K in {2048, 768, 64}, all divisible
    stage(0, 0);
    for (int c = 0; c < nch; ++c) {
        asm volatile("s_wait_asynccnt 0x0" ::: "memory");   // staging for chunk c done
        __syncthreads();
        if (c + 1 < nch) stage((c + 1) & 1, (c + 1) * KC);  // overlap next chunk with compute
        compute(c & 1, c * KC);
    }

    if (!active) return;
    float bb0 = 0.f, bb1 = 0.f, bb2 = 0.f, bb3 = 0.f;
    if (bias) { bb0 = bias[r]; bb1 = bias[r + 16]; bb2 = bias[r + 32]; bb3 = bias[r + 48]; }
    const int rowoff = half * 8;
#pragma unroll
    for (int rr = 0; rr < 8; ++rr) {
        float* crow = C + (size_t)(m0 + rowoff + rr) * ldc;
        crow[r]      = acc0[rr] + bb0;
        crow[r + 16] = acc1[rr] + bb1;
        crow[r + 32] = acc2[rr] + bb2;
        crow[r + 48] = acc3[rr] + bb3;
    }
}

// ---------- ACC lives in d_out in final (permuted) layout ----------
__device__ __forceinline__ size_t acc_index(int n, int j) {
    int c = j >> 6;                 // 0:g 1:v 2:t
    int jj = j & 63;
    int item = (n >= NUM_USER) ? 1 : 0;
    int nl = n - item * NUM_USER;
    return ((size_t)(2 * c + item) * NUM_USER + nl) * 64 + jj;
}

__global__ void k_acc_init(const float* __restrict__ X, float* __restrict__ out) {
    int idx = blockIdx.x * blockDim.x + threadIdx.x;    // NN*FDIM threads
    if (idx >= NN * FDIM) return;
    int n = idx / FDIM, j = idx % FDIM;
    out[acc_index(n, j)] = X[idx];
}

// fold self-loop into propagation target: X <- H * dinv[n]^2  (X is consumed, reused as Y)
__global__ void k_selfloop(const float* __restrict__ H, const float* __restrict__ dinv,
                           float* __restrict__ X) {
    int idx = blockIdx.x * blockDim.x + threadIdx.x;
    if (idx >= NN * FDIM) return;
    int n = idx / FDIM;
    float d = dinv[n];
    X[idx] = H[idx] * d * d;
}

// one wave per (undirected) edge: scatter both directions, 192 cols = 6 floats/lane.
// H and Y (153MB total) are L2-resident on MI455X (192MB), so atomics run at L2 rate.
__global__ void k_propagate(const int* __restrict__ edges, const float* __restrict__ dinv,
                            const float* __restrict__ H, float* __restrict__ Y, int e) {
    int w = (blockIdx.x * blockDim.x + threadIdx.x) >> 5;
    int lane = threadIdx.x & 31;
    if (w >= e) return;
    int u  = edges[2 * w];
    int it = edges[2 * w + 1] + NUM_USER;
    float norm = dinv[u] * dinv[it];
    const float* hu = H + (size_t)u * FDIM;
    const float* hi = H + (size_t)it * FDIM;
    float* yu = Y + (size_t)u * FDIM;
    float* yi = Y + (size_t)it * FDIM;
#pragma unroll
    for (int j = 0; j < 6; ++j) {
        int col = j * 32 + lane;
        atomicAdd(yi + col, hu[col] * norm);
        atomicAdd(yu + col, hi[col] * norm);
    }
}

// X <- X + bias ; ACC(out) += X
__global__ void k_post(float* __restrict__ X, const float* __restrict__ bias,
                       float* __restrict__ out) {
    int idx = blockIdx.x * blockDim.x + threadIdx.x;
    if (idx >= NN * FDIM) return;
    int n = idx / FDIM, j = idx % FDIM;
    float x = X[idx] + bias[j & 63];
    X[idx] = x;
    out[acc_index(n, j)] += x;
}

__global__ void k_scale(float* __restrict__ out, int n) {
    int i = blockIdx.x * blockDim.x + threadIdx.x;
    if (i < n) out[i] *= (1.0f / 3.0f);
}

// ---------- host ----------
extern "C" void kernel_launch(void* const* d_in, const int* in_sizes, int n_in,
                              void* d_out, int out_size, void* d_ws, size_t ws_size,
                              hipStream_t stream) {
    const int*   edges    = (const int*)  d_in[0];
    const float* v_feat   = (const float*)d_in[1];
    const float* t_feat   = (const float*)d_in[2];
    const float* user_emb = (const float*)d_in[3];
    const float* item_emb = (const float*)d_in[4];
    const float* user_v   = (const float*)d_in[5];
    const float* user_t   = (const float*)d_in[6];
    const float* W_v      = (const float*)d_in[7];
    const float* b_v      = (const float*)d_in[8];
    const float* W_t      = (const float*)d_in[9];
    const float* b_t      = (const float*)d_in[10];
    const float* Wl[2]    = { (const float*)d_in[11], (const float*)d_in[13] };
    const float* bl[2]    = { (const float*)d_in[12], (const float*)d_in[14] };
    float* out = (float*)d_out;

    float* dinv = (float*)d_ws;                        // NN floats
    float* X    = dinv + NN;                           // NN*FDIM (also reused as Y)
    float* H    = X + (size_t)NN * FDIM;               // NN*FDIM

    const int T = 256;
    const int gN   = (NN + T - 1) / T;
    const int gE   = (NEDGE + T - 1) / T;
    const int gU64 = (NUM_USER * 64 + T - 1) / T;
    const int gNF  = (NN * FDIM + T - 1) / T;
    const int gProp = (NEDGE * 32 + T - 1) / T;        // one wave per edge

    // graph normalization
    k_fill1<<<gN, T, 0, stream>>>(dinv, NN);
    k_deg_edges<<<gE, T, 0, stream>>>(edges, dinv, NEDGE);
    k_rsqrt<<<gN, T, 0, stream>>>(dinv, NN);

    // X0 = [g | v | t], item v/t blocks via WMMA feature GEMMs
    k_init_users<<<gU64, T, 0, stream>>>(user_emb, user_v, user_t, X);
    k_init_items_g<<<gU64, T, 0, stream>>>(item_emb, X);
    {
        int grid = (NUM_USER + 127) / 128;
        k_gemm16x64<<<grid, T, 0, stream>>>(v_feat, 2048, W_v, 2048, b_v,
                                            X + (size_t)NUM_USER * FDIM + 64, FDIM, NUM_USER);
        k_gemm16x64<<<grid, T, 0, stream>>>(t_feat, 768, W_t, 768, b_t,
                                            X + (size_t)NUM_USER * FDIM + 128, FDIM, NUM_USER);
    }
    k_acc_init<<<gNF, T, 0, stream>>>(X, out);

    // two fused GCN layers over all three stacks at once
    for (int l = 0; l < 2; ++l) {
        int grid = (NN + 127) / 128;
        for (int c = 0; c < 3; ++c)
            k_gemm16x64<<<grid, T, 0, stream>>>(X + 64 * c, FDIM, Wl[l], 64, nullptr,
                                                H + 64 * c, FDIM, NN);
        k_selfloop<<<gNF, T, 0, stream>>>(H, dinv, X);          // X becomes Y (self-loop seeded)
        k_propagate<<<gProp, T, 0, stream>>>(edges, dinv, H, X, NEDGE);
        k_post<<<gNF, T, 0, stream>>>(X, bl[l], out);
    }
    k_scale<<<gNF, T, 0, stream>>>(out, NN * FDIM);
}